// Analyzer_45775761440988
// MI455X (gfx1250) — compile-verified
//
#include <hip/hip_runtime.h>
#include <hip/hip_bf16.h>

// ---------------------------------------------------------------------------
// Analyzer (NMS + TP matching) for MI455X / gfx1250.
// Distance tiles via V_WMMA_F32_16X16X4_F32 with all predicates folded into
// the operands:
//   A (16x4) row i : ( ax, ay, az, |a|^2 [+ 1e30 if row suppressed-gate] )
//   B (4x16) col j : (-2bx,-2by,-2bz, 1 )
//   C (16x16)      : |b_j|^2 - d_thr^2   (+inf if column invalid)
// => D[i][j] = dist^2 - d_thr^2 (biased), so "neighbor" == sign bit of D.
// The sorted positions are pre-swizzled into the WMMA A-operand lane layout
// (sA), so the NMS inner loop is: b64 load -> wmma -> 8x sign-extract.
// ---------------------------------------------------------------------------

namespace {
constexpr int kB = 2, kC = 32, kZ = 4, kE = 2;
constexpr int kN  = kC * kC * kZ;   // 4096 points per (b,e)
constexpr int kG  = kB * kE;        // 4 (b,e) pairs
constexpr int kNT = kN / 16;        // 256 16-wide tiles per dim

// d_out layout (float32), concatenated in reference return order
constexpr int OFF_PP   = 0;                    // p_pos       [G,N,3]
constexpr int OFF_MP   = OFF_PP   + kG*kN*3;   // mask_p      [G,N]
constexpr int OFF_KEEP = OFF_MP   + kG*kN;     // keep_nms    [G,N]
constexpr int OFF_TPOS = OFF_KEEP + kG*kN;     // t_pos       [G,N,3]
constexpr int OFF_MT   = OFF_TPOS + kG*kN*3;   // mask_t      [G,N]
constexpr int OFF_TPT  = OFF_MT   + kG*kN;     // tp_t_mask   [G,N]
constexpr int OFF_TPP  = OFF_TPT  + kG*kN;     // tp_p_idx    [G,N]
constexpr int OFF_TPTN = OFF_TPP  + kG*kN;     // tp_t_mask_nms
constexpr int OFF_TPPN = OFF_TPTN + kG*kN;     // tp_p_idx_nms
}  // namespace

typedef __attribute__((ext_vector_type(2))) float v2f;
typedef __attribute__((ext_vector_type(8))) float v8f;

__device__ __forceinline__ float ele_d2_of(int e) {
    return (e == 0) ? (0.74f * 0.74f) : (0.528f * 0.528f);
}

__device__ __forceinline__ unsigned sgn(float x) {   // 1 iff x < 0 (or -0)
    return __float_as_uint(x) >> 31;
}

__device__ __forceinline__ v8f wmma_tile(v2f Av, v2f Bv, v8f Cv) {
    return __builtin_amdgcn_wmma_f32_16x16x4_f32(
        false, Av, false, Bv, (short)0, Cv, false, false);
}

// ---------------------------------------------------------------------------
// Kernel 1: positions / confidences / masks; write 4 direct outputs and stage
// (x,y,z,|p|^2) float4s for the WMMA passes.
// ---------------------------------------------------------------------------
__global__ void k_prep(const float* __restrict__ pred,
                       const float* __restrict__ tgt,
                       float* __restrict__ out,
                       float* __restrict__ p4, float* __restrict__ pconf,
                       int* __restrict__ pvalid,
                       float* __restrict__ t4, int* __restrict__ tvalid) {
    int gid = blockIdx.x * blockDim.x + threadIdx.x;   // over G*N
    if (gid >= kG * kN) return;
    int g = gid / kN, n = gid % kN;
    int b = g / kE, e = g % kE;
    int z  = n % kZ;
    int c2 = (n / kZ) % kC;
    int c1 = n / (kZ * kC);

    // input element [b][c1][c2][z][e][k]
    size_t base = (((((size_t)b * kC + c1) * kC + c2) * kZ + z) * kE + e) * 4;
    const float lx = 25.0f / kC, ly = 25.0f / kC, lz = 3.0f / kZ;

    float px = (pred[base + 0] + (float)c1) * lx;
    float py = (pred[base + 1] + (float)c2) * ly;
    float pz = (pred[base + 2] + (float)z ) * lz;
    float pc = pred[base + 3];
    float tx = (tgt[base + 0] + (float)c1) * lx;
    float ty = (tgt[base + 1] + (float)c2) * ly;
    float tz = (tgt[base + 2] + (float)z ) * lz;
    float tc = tgt[base + 3];

    int mp = pc > 0.5f;
    int mt = tc > 0.5f;

    // direct outputs
    out[OFF_PP   + gid * 3 + 0] = px;
    out[OFF_PP   + gid * 3 + 1] = py;
    out[OFF_PP   + gid * 3 + 2] = pz;
    out[OFF_MP   + gid] = mp ? 1.0f : 0.0f;
    out[OFF_TPOS + gid * 3 + 0] = tx;
    out[OFF_TPOS + gid * 3 + 1] = ty;
    out[OFF_TPOS + gid * 3 + 2] = tz;
    out[OFF_MT   + gid] = mt ? 1.0f : 0.0f;

    // staging
    p4[gid * 4 + 0] = px; p4[gid * 4 + 1] = py;
    p4[gid * 4 + 2] = pz; p4[gid * 4 + 3] = px*px + py*py + pz*pz;
    pconf[gid]  = pc;
    pvalid[gid] = mp;
    t4[gid * 4 + 0] = tx; t4[gid * 4 + 1] = ty;
    t4[gid * 4 + 2] = tz; t4[gid * 4 + 3] = 0.0f;
    tvalid[gid] = mt;
}

// ---------------------------------------------------------------------------
// Kernel 2: per-(b,e) bitonic argsort in LDS.
// key = valid ? -conf : +inf, ascending  ==  confidence-descending with all
// invalid points compacted to the tail  ->  sorted validity is (j < n_valid).
// Emits two sorted-position views:
//   s4 : float4 (x,y,z,|p|^2) per sorted row      (B/C operand construction)
//   sA : WMMA A-operand lane layout, float2 per (tile,lane):
//        lane<16 -> (ax,ay) of row lane; lane>=16 -> (az,|a|^2) of row lane-16
// ---------------------------------------------------------------------------
__global__ void __launch_bounds__(1024)
k_sort(const float* __restrict__ pconf, const int* __restrict__ pvalid,
       const float* __restrict__ p4,
       int* __restrict__ ord, float* __restrict__ s4, float* __restrict__ sA,
       int* __restrict__ nval) {
    __shared__ float key[kN];
    __shared__ int   idx[kN];
    __shared__ int   s_cnt;
    const int g = blockIdx.x;
    const int t = threadIdx.x;

    for (int i = t; i < kN; i += 1024) {
        key[i] = pvalid[g * kN + i] ? -pconf[g * kN + i] : __builtin_inff();
        idx[i] = i;
    }
    if (t == 0) s_cnt = 0;
    __syncthreads();

    for (int k = 2; k <= kN; k <<= 1) {
        for (int j = k >> 1; j > 0; j >>= 1) {
            for (int i = t; i < kN; i += 1024) {
                int ixj = i ^ j;
                if (ixj > i) {
                    bool up = ((i & k) == 0);
                    float a = key[i], c = key[ixj];
                    if (up ? (a > c) : (a < c)) {
                        key[i] = c; key[ixj] = a;
                        int tmp = idx[i]; idx[i] = idx[ixj]; idx[ixj] = tmp;
                    }
                }
            }
            __syncthreads();
        }
    }

    float2* sA2 = (float2*)sA;
    int local = 0;
    for (int i = t; i < kN; i += 1024) {
        int o = idx[i];
        ord[g * kN + i] = o;
        float4 v = ((const float4*)p4)[g * kN + o];
        ((float4*)s4)[g * kN + i] = v;
        int tile = i >> 4, l = i & 15;
        sA2[g * kN * 2 + tile * 32 + l]      = make_float2(v.x, v.y); // K0,K1
        sA2[g * kN * 2 + tile * 32 + 16 + l] = make_float2(v.z, v.w); // K2,K3
        local += pvalid[g * kN + i];
    }
    atomicAdd(&s_cnt, local);
    __syncthreads();
    if (t == 0) nval[g] = s_cnt;
}

// ---------------------------------------------------------------------------
// Kernel 3: NMS pass 1 — restrain[j] = #(i<j, both valid, dist<d).
// One wave per (g, column-tile J).  A operands are pre-swizzled (1 b64 load),
// threshold & column validity live in C, so interior tiles are pure
// wmma + sign-bit popcount.
// ---------------------------------------------------------------------------
__global__ void __launch_bounds__(32)
k_restrain(const float* __restrict__ s4, const float* __restrict__ sA,
           const int* __restrict__ nval, unsigned* __restrict__ restr) {
    const int wid = blockIdx.x;           // G*NT waves
    const int g = wid / kNT, J = wid % kNT;
    const float dd = ele_d2_of(g % kE);
    const int lane = threadIdx.x, half = lane >> 4, l15 = lane & 15;
    const int nv = nval[g];
    const float4* s4g  = ((const float4*)s4) + g * kN;
    const float2* sAg  = ((const float2*)sA) + g * kN * 2;

    float4 bp = s4g[J * 16 + l15];
    v2f Bv;
    Bv[0] = half ? (-2.0f * bp.z) : (-2.0f * bp.x);
    Bv[1] = half ? 1.0f           : (-2.0f * bp.y);
    const int gj = J * 16 + l15;
    const float cb = (gj < nv) ? (bp.w - dd) : __builtin_inff();
    v8f Cv;
#pragma unroll
    for (int r = 0; r < 8; ++r) Cv[r] = cb;

    unsigned count = 0;
    for (int I = 0; I < J; ++I) {               // interior: i<j guaranteed
        float2 a = sAg[I * 32 + lane];
        v2f Av; Av[0] = a.x; Av[1] = a.y;
        v8f D = wmma_tile(Av, Bv, Cv);
#pragma unroll
        for (int r = 0; r < 8; ++r) count += sgn(D[r]);
    }
    {                                            // diagonal tile: i<j per-row
        float2 a = sAg[J * 32 + lane];
        v2f Av; Av[0] = a.x; Av[1] = a.y;
        v8f D = wmma_tile(Av, Bv, Cv);
        const int rmax = l15 - half * 8;         // rows r<rmax are above diag
#pragma unroll
        for (int r = 0; r < 8; ++r) count += (r < rmax) ? sgn(D[r]) : 0u;
    }
    count += __shfl_xor((int)count, 16, 32);     // combine row halves
    if (half == 0) restr[g * kN + gj] = count;
}

// ---------------------------------------------------------------------------
// Kernel 4: NMS pass 2 — correct[j] = sum_i (restrain[i]!=0) * adj[i][j];
// rows with restrain==0 are knocked out via a +1e30 bias on the |a|^2 term
// (held in the upper half-wave's Av[1]).
// ---------------------------------------------------------------------------
__global__ void __launch_bounds__(32)
k_keep(const float* __restrict__ s4, const float* __restrict__ sA,
       const int* __restrict__ nval, const unsigned* __restrict__ restr,
       const int* __restrict__ ord, float* __restrict__ out,
       int* __restrict__ keepm) {
    const int wid = blockIdx.x;
    const int g = wid / kNT, J = wid % kNT;
    const float dd = ele_d2_of(g % kE);
    const int lane = threadIdx.x, half = lane >> 4, l15 = lane & 15;
    const int nv = nval[g];
    const float4* s4g = ((const float4*)s4) + g * kN;
    const float2* sAg = ((const float2*)sA) + g * kN * 2;
    const unsigned* rg = restr + g * kN;

    float4 bp = s4g[J * 16 + l15];
    v2f Bv;
    Bv[0] = half ? (-2.0f * bp.z) : (-2.0f * bp.x);
    Bv[1] = half ? 1.0f           : (-2.0f * bp.y);
    const int gj = J * 16 + l15;
    const float cb = (gj < nv) ? (bp.w - dd) : __builtin_inff();
    v8f Cv;
#pragma unroll
    for (int r = 0; r < 8; ++r) Cv[r] = cb;

    unsigned correct = 0;
    for (int I = 0; I < J; ++I) {
        float2 a = sAg[I * 32 + lane];
        // restrain_one gate: +1e30 on |a|^2 (upper half lanes only)
        float wb = (half && rg[I * 16 + l15] == 0u) ? 1e30f : 0.0f;
        v2f Av; Av[0] = a.x; Av[1] = a.y + wb;
        v8f D = wmma_tile(Av, Bv, Cv);
#pragma unroll
        for (int r = 0; r < 8; ++r) correct += sgn(D[r]);
    }
    {
        float2 a = sAg[J * 32 + lane];
        float wb = (half && rg[J * 16 + l15] == 0u) ? 1e30f : 0.0f;
        v2f Av; Av[0] = a.x; Av[1] = a.y + wb;
        v8f D = wmma_tile(Av, Bv, Cv);
        const int rmax = l15 - half * 8;
#pragma unroll
        for (int r = 0; r < 8; ++r) correct += (r < rmax) ? sgn(D[r]) : 0u;
    }
    correct += __shfl_xor((int)correct, 16, 32);
    if (half == 0) {
        bool keep = ((int)rg[gj] - (int)correct == 0) && (gj < nv);
        int o = ord[g * kN + gj];              // permutation -> full coverage
        keepm[g * kN + o] = keep ? 1 : 0;
        out[OFF_KEEP + g * kN + o] = keep ? 1.0f : 0.0f;
    }
}

// ---------------------------------------------------------------------------
// Kernel 5: TP matching (mask_p and keep_nms variants in one pass).
// P points cached in LDS; each thread owns one T row; running (min, argmin,
// any) trackers preserve jnp.argmin first-index semantics (strict <).
// ---------------------------------------------------------------------------
__global__ void __launch_bounds__(256)
k_match(const float* __restrict__ t4, const int* __restrict__ tvalid,
        const float* __restrict__ p4, const int* __restrict__ pvalid,
        const int* __restrict__ keepm, float* __restrict__ out) {
    __shared__ float    sp[kN * 3];          // 48 KB of the 320 KB WGP LDS
    __shared__ unsigned smask[kN / 32];
    __shared__ unsigned skeep[kN / 32];

    const int blk = blockIdx.x;
    const int g = blk / 16, sub = blk % 16;
    const float dd = ele_d2_of(g % kE);
    const int t = threadIdx.x;

    for (int i = t; i < kN; i += 256) {
        float4 v = ((const float4*)p4)[g * kN + i];
        sp[i * 3 + 0] = v.x; sp[i * 3 + 1] = v.y; sp[i * 3 + 2] = v.z;
    }
    for (int i = t; i < kN / 32; i += 256) {
        unsigned m = 0, k = 0;
        for (int b = 0; b < 32; ++b) {
            m |= (pvalid[g * kN + i * 32 + b] ? 1u : 0u) << b;
            k |= (keepm [g * kN + i * 32 + b] ? 1u : 0u) << b;
        }
        smask[i] = m; skeep[i] = k;
    }
    __syncthreads();

    const int row = sub * 256 + t;
    float4 tp = ((const float4*)t4)[g * kN + row];
    const bool tv = tvalid[g * kN + row] != 0;
    const float inf = __builtin_inff();

    float best1 = inf, best2 = inf;
    int   i1 = 0, i2 = 0;
    bool  any1 = false, any2 = false;
    for (int j = 0; j < kN; ++j) {
        float dx = tp.x - sp[j * 3 + 0];
        float dy = tp.y - sp[j * 3 + 1];
        float dz = tp.z - sp[j * 3 + 2];
        float d2 = dx * dx + dy * dy + dz * dz;
        bool pm = (smask[j >> 5] >> (j & 31)) & 1u;
        bool pk = (skeep[j >> 5] >> (j & 31)) & 1u;
        float d2a = (tv && pm) ? d2 : inf;   // dist<d  <=>  d2<d^2 (monotone)
        float d2b = (tv && pk) ? d2 : inf;
        any1 |= (d2a < dd);
        any2 |= (d2b < dd);
        if (d2a < best1) { best1 = d2a; i1 = j; }
        if (d2b < best2) { best2 = d2b; i2 = j; }
    }
    out[OFF_TPT  + g * kN + row] = any1 ? 1.0f : 0.0f;
    out[OFF_TPP  + g * kN + row] = (float)i1;
    out[OFF_TPTN + g * kN + row] = any2 ? 1.0f : 0.0f;
    out[OFF_TPPN + g * kN + row] = (float)i2;
}

// ---------------------------------------------------------------------------
extern "C" void kernel_launch(void* const* d_in, const int* in_sizes, int n_in,
                              void* d_out, int out_size, void* d_ws, size_t ws_size,
                              hipStream_t stream) {
    const float* pred = (const float*)d_in[0];
    const float* tgt  = (const float*)d_in[1];
    float* out = (float*)d_out;

    // workspace carve-up (~1.35 MB, 16B-aligned blocks)
    float*    p4    = (float*)d_ws;              // G*N*4
    float*    pconf = p4 + kG * kN * 4;          // G*N
    int*      pval  = (int*)(pconf + kG * kN);   // G*N
    float*    t4    = (float*)(pval + kG * kN);  // G*N*4
    int*      tval  = (int*)(t4 + kG * kN * 4);  // G*N
    int*      ord   = tval + kG * kN;            // G*N
    float*    s4    = (float*)(ord + kG * kN);   // G*N*4
    int*      nval  = (int*)(s4 + kG * kN * 4);  // 16 (padded)
    unsigned* restr = (unsigned*)(nval + 16);    // G*N
    int*      keepm = (int*)(restr + kG * kN);   // G*N
    float*    sA    = (float*)(keepm + kG * kN); // G*N*4 (A-operand swizzle)

    k_prep    <<<(kG * kN) / 256, 256, 0, stream>>>(pred, tgt, out, p4, pconf,
                                                    pval, t4, tval);
    k_sort    <<<kG, 1024, 0, stream>>>(pconf, pval, p4, ord, s4, sA, nval);
    k_restrain<<<kG * kNT, 32, 0, stream>>>(s4, sA, nval, restr);
    k_keep    <<<kG * kNT, 32, 0, stream>>>(s4, sA, nval, restr, ord, out, keepm);
    k_match   <<<kG * 16, 256, 0, stream>>>(t4, tval, p4, pval, keepm, out);
}